// GIN_86629490360414
// MI455X (gfx1250) — compile-verified
//
#include <hip/hip_runtime.h>

// GIN forward: 3x GINConv(eps=0, MLP 128->128->128, relu) + head 128->128->64.
// N=50000 nodes, E=800000 edges, D=128. Working set (~60MB) is L2-resident on
// MI455X (192MB L2) => scatter-add bound; GEMMs (~12 GFLOP) use exact fp32
// WMMA (v_wmma_f32_16x16x4_f32, confirmed lowering in round 1).

typedef float v2f __attribute__((ext_vector_type(2)));
typedef float v8f __attribute__((ext_vector_type(8)));

#define DH 128

static __device__ __forceinline__ v8f zero8() {
  v8f z = {0.f, 0.f, 0.f, 0.f, 0.f, 0.f, 0.f, 0.f};
  return z;
}

// agg[dst] += h[src] over all edges; one wave per edge, lane handles 4 floats.
__global__ void __launch_bounds__(256) gin_scatter_add(
    const float* __restrict__ h, const int* __restrict__ src,
    const int* __restrict__ dst, float* __restrict__ agg, int E) {
  const int lane = threadIdx.x & 31;
  int w = (int)((blockIdx.x * blockDim.x + threadIdx.x) >> 5);
  const int nw = (int)((gridDim.x * blockDim.x) >> 5);
  for (int e = w; e < E; e += nw) {
    const int s = src[e];
    const int d = dst[e];
    const float4 v =
        *reinterpret_cast<const float4*>(h + (size_t)s * DH + lane * 4);
    float* p = agg + (size_t)d * DH + lane * 4;
    atomicAdd(p + 0, v.x);
    atomicAdd(p + 1, v.y);
    atomicAdd(p + 2, v.z);
    atomicAdd(p + 3, v.w);
  }
}

// out[row, 0:NCOL2*16) = act2( relu( (h [+ agg]) @ W1 + b1 ) @ W2 + b2 )
// W1: [128,128] row-major, W2: [128, NCOL2*16] row-major.
// LDS weight layout is pair-interleaved over K: sW[(k>>1)*SWP + n] is the
// float2 (W[k][n], W[k+1][n]) -> one aligned ds_load_b64 per B fragment,
// no repacking movs. SWP=144 => lanes 16-31 sit 32 banks from lanes 0-15.
template <int NCOL2, bool HAS_AGG, bool RELU2>
__global__ void __launch_bounds__(128) gin_mlp(
    const float* __restrict__ hin, const float* __restrict__ agg,
    const float* __restrict__ W1, const float* __restrict__ b1,
    const float* __restrict__ W2, const float* __restrict__ b2,
    float* __restrict__ out, int nrows) {
  constexpr int N2 = NCOL2 * 16;
  constexpr int SWP = 144;  // pair-row stride in float2 units
  constexpr int SZ = 132;   // padded row stride for staged activations
  __shared__ v2f sW1[(DH / 2) * SWP];
  __shared__ v2f sW2[(DH / 2) * SWP];
  __shared__ float sb1[DH];
  __shared__ float sb2[DH];
  __shared__ float sZ[4][16 * SZ];

  const int tid = (int)threadIdx.x;

  // Cooperative weight/bias load into LDS (pair-interleaved layout).
  for (int i = tid; i < DH * DH; i += 128) {
    const int r = i >> 7, c = i & 127;
    reinterpret_cast<float*>(sW1)[((r >> 1) * SWP + c) * 2 + (r & 1)] = W1[i];
  }
  for (int i = tid; i < DH * N2; i += 128) {
    const int r = i / N2, c = i - r * N2;
    reinterpret_cast<float*>(sW2)[((r >> 1) * SWP + c) * 2 + (r & 1)] = W2[i];
  }
  if (tid < DH) sb1[tid] = b1[tid];
  if (tid < N2) sb2[tid] = b2[tid];
  __syncthreads();

  const int lane = tid & 31;
  const int wid = tid >> 5;
  const int m = lane & 15;           // A: row in tile; B/C: col in tile
  const int koff = (lane >> 4) * 2;  // lanes 0-15 -> K 0,1; 16-31 -> K 2,3
  const int kpo = lane >> 4;         // pair-row offset for this half-wave
  const int mrow = (lane >> 4) * 8;  // C layout: lanes 16-31 hold rows 8..15
  float* zs = sZ[wid];

  const int ntiles = (nrows + 15) >> 4;
  const int gw0 = (int)blockIdx.x * 4 + wid;
  const int gwn = (int)gridDim.x * 4;

  for (int t = gw0; t < ntiles; t += gwn) {
    const int row0 = t << 4;
    int rowa = row0 + m;
    if (rowa > nrows - 1) rowa = nrows - 1;  // clamp tail loads, guard stores
    const float* hrow = hin + (size_t)rowa * DH;
    const float* arow = agg + (size_t)rowa * DH;  // unused unless HAS_AGG

    v8f acc[8];
#pragma unroll
    for (int n = 0; n < 8; ++n) acc[n] = zero8();

    // ---- GEMM1: z = (h [+ agg]) @ W1 ----
    for (int k0 = 0; k0 < DH; k0 += 4) {
      const int k = k0 + koff;
      const int kp = (k0 >> 1) + kpo;
      v2f a = *reinterpret_cast<const v2f*>(hrow + k);
      if (HAS_AGG) {
        const v2f av = *reinterpret_cast<const v2f*>(arow + k);
        a.x += av.x;
        a.y += av.y;
      }
#pragma unroll
      for (int n = 0; n < 8; ++n) {
        const v2f bf = sW1[kp * SWP + n * 16 + m];
        acc[n] = __builtin_amdgcn_wmma_f32_16x16x4_f32(
            false, a, false, bf, (short)0, acc[n], false, false);
      }
    }

    // bias + relu, stage 16x128 tile row-major into per-wave LDS slice
#pragma unroll
    for (int n = 0; n < 8; ++n) {
      const float bias = sb1[n * 16 + m];
#pragma unroll
      for (int j = 0; j < 8; ++j) {
        float v = acc[n][j] + bias;
        v = v > 0.0f ? v : 0.0f;
        zs[(mrow + j) * SZ + n * 16 + m] = v;
      }
    }
    __builtin_amdgcn_wave_barrier();

    // ---- GEMM2: out = z @ W2 ----
#pragma unroll
    for (int n = 0; n < NCOL2; ++n) acc[n] = zero8();
    for (int k0 = 0; k0 < DH; k0 += 4) {
      const int k = k0 + koff;
      const int kp = (k0 >> 1) + kpo;
      const v2f a = *reinterpret_cast<const v2f*>(&zs[m * SZ + k]);
#pragma unroll
      for (int n = 0; n < NCOL2; ++n) {
        const v2f bf = sW2[kp * SWP + n * 16 + m];
        acc[n] = __builtin_amdgcn_wmma_f32_16x16x4_f32(
            false, a, false, bf, (short)0, acc[n], false, false);
      }
    }
    __builtin_amdgcn_wave_barrier();

    // epilogue: bias (+ optional relu), guarded store
#pragma unroll
    for (int n = 0; n < NCOL2; ++n) {
      const float bias = sb2[n * 16 + m];
#pragma unroll
      for (int j = 0; j < 8; ++j) {
        float v = acc[n][j] + bias;
        if (RELU2) v = v > 0.0f ? v : 0.0f;
        const int row = row0 + mrow + j;
        if (row < nrows) out[(size_t)row * N2 + n * 16 + m] = v;
      }
    }
  }
}

extern "C" void kernel_launch(void* const* d_in, const int* in_sizes, int n_in,
                              void* d_out, int out_size, void* d_ws,
                              size_t ws_size, hipStream_t stream) {
  const float* x = (const float*)d_in[0];
  const int* ei = (const int*)d_in[1];
  const int N = in_sizes[0] / DH;
  const int E = in_sizes[1] / 2;
  const int* src = ei;
  const int* dst = ei + E;

  const float* c1w1 = (const float*)d_in[2];
  const float* c1b1 = (const float*)d_in[3];
  const float* c1w2 = (const float*)d_in[4];
  const float* c1b2 = (const float*)d_in[5];
  const float* c2w1 = (const float*)d_in[6];
  const float* c2b1 = (const float*)d_in[7];
  const float* c2w2 = (const float*)d_in[8];
  const float* c2b2 = (const float*)d_in[9];
  const float* c3w1 = (const float*)d_in[10];
  const float* c3b1 = (const float*)d_in[11];
  const float* c3w2 = (const float*)d_in[12];
  const float* c3b2 = (const float*)d_in[13];
  const float* l1w = (const float*)d_in[14];
  const float* l1b = (const float*)d_in[15];
  const float* l2w = (const float*)d_in[16];
  const float* l2b = (const float*)d_in[17];

  float* hA = (float*)d_ws;
  float* hB = hA + (size_t)N * DH;
  float* agg = hB + (size_t)N * DH;
  const size_t featBytes = (size_t)N * DH * sizeof(float);

  const dim3 sB(256), sG(4096);
  const dim3 mB(128), mG(512);

  // layer 1
  hipMemsetAsync(agg, 0, featBytes, stream);
  gin_scatter_add<<<sG, sB, 0, stream>>>(x, src, dst, agg, E);
  gin_mlp<8, true, true>
      <<<mG, mB, 0, stream>>>(x, agg, c1w1, c1b1, c1w2, c1b2, hA, N);
  // layer 2
  hipMemsetAsync(agg, 0, featBytes, stream);
  gin_scatter_add<<<sG, sB, 0, stream>>>(hA, src, dst, agg, E);
  gin_mlp<8, true, true>
      <<<mG, mB, 0, stream>>>(hA, agg, c2w1, c2b1, c2w2, c2b2, hB, N);
  // layer 3
  hipMemsetAsync(agg, 0, featBytes, stream);
  gin_scatter_add<<<sG, sB, 0, stream>>>(hB, src, dst, agg, E);
  gin_mlp<8, true, true>
      <<<mG, mB, 0, stream>>>(hB, agg, c3w1, c3b1, c3w2, c3b2, hA, N);
  // head: relu(hA @ l1w + l1b) @ l2w + l2b -> [N, 64]
  gin_mlp<4, false, false>
      <<<mG, mB, 0, stream>>>(hA, nullptr, l1w, l1b, l2w, l2b,
                              (float*)d_out, N);
}